// CSPN_25374666784912
// MI455X (gfx1250) — compile-verified
//
#include <hip/hip_runtime.h>
#include <hip/hip_bf16.h>
#include <stdint.h>

// Problem constants (fixed by the reference)
#define KSZ 5
#define RAD 2
#define BATCH 4
#define H 352
#define W 1216
#define HP (H + KSZ - 1)          // 356
#define WP (W + KSZ - 1)          // 1220
#define GW_HW (HP * WP)           // 434320 elements per (b, tap) plane

// Tiling: 1216 = 19*64, 352 = 22*16 -> exact tiling, no ragged edges
#define TW 64
#define TH 16
#define LDSW (TW + 4)             // 68
#define LDSH (TH + 4)             // 20
#define TILE_ELEMS (LDSW * LDSH)  // 1360 floats = 5440 B LDS
#define NTHREADS 256              // 8 wave32 waves

__global__ __launch_bounds__(NTHREADS)
void cspn_25tap_kernel(const float* __restrict__ gw,
                       const float* __restrict__ hn,
                       const float* __restrict__ h0,
                       float* __restrict__ out)
{
    __shared__ float tile[TILE_ELEMS];

    const int tid = threadIdx.x;
    const int bx0 = blockIdx.x * TW;
    const int by0 = blockIdx.y * TH;
    const int b   = blockIdx.z;

    // ---- Phase 1: zero-fill the halo tile with b128 stores ----
    // (covers the zero-padding of hn at image borders)
    {
        float4* t4 = (float4*)tile;
        const float4 z = make_float4(0.f, 0.f, 0.f, 0.f);
        for (int i = tid; i < TILE_ELEMS / 4; i += NTHREADS)   // 340 stores
            t4[i] = z;
    }
    __syncthreads();

    // ---- Phase 2: async global->LDS fill of the hn tile (CDNA5 ASYNC path) ----
    // tile[r][c] <-> hn[b][by0 + r - 2][bx0 + c - 2]; OOB lanes stay EXEC-masked,
    // leaving the zeros from phase 1.
    {
        const float* hnb = hn + (size_t)b * ((size_t)H * W);
        for (int i = tid; i < TILE_ELEMS; i += NTHREADS) {
            const int r  = i / LDSW;
            const int c  = i - r * LDSW;
            const int gy = by0 + r - RAD;
            const int gx = bx0 + c - RAD;
            if ((unsigned)gy < (unsigned)H && (unsigned)gx < (unsigned)W) {
                // LDS aperture: low 32 bits of a generic shared pointer are the
                // byte offset from LDS_BASE (ISA 10.2 aperture mapping).
                uint32_t lds_addr = (uint32_t)(uintptr_t)(&tile[i]);
                uint64_t gaddr    = (uint64_t)(uintptr_t)(hnb + (size_t)gy * W + gx);
                asm volatile("global_load_async_to_lds_b32 %0, %1, off"
                             :: "v"(lds_addr), "v"(gaddr)
                             : "memory");
            }
        }
        // Drain this wave's async copies, then make LDS visible block-wide.
        asm volatile("s_wait_asynccnt 0x0" ::: "memory");
    }
    __syncthreads();

    // ---- Phase 3: 25-tap weighted sum, 4 pixels per thread ----
    const int lx  = tid & (TW - 1);   // 0..63
    const int ly0 = tid >> 6;         // 0..3

    const float* __restrict__ gwb  = gw + (size_t)b * 25u * (size_t)GW_HW;
    const float* __restrict__ h0b  = h0 + (size_t)b * ((size_t)H * W);
    float*       __restrict__ outb = out + (size_t)b * ((size_t)H * W);

    #pragma unroll
    for (int rr = 0; rr < TH / 4; ++rr) {
        const int ly = ly0 + rr * 4;
        const int y  = by0 + ly;
        const int x  = bx0 + lx;

        const float h0v = h0b[(size_t)y * W + x];
        // Per-thread part of the weight address: 32-bit VGPR offset
        // (max (353*1220+1217)*4 B ~ 1.73 MB, fits comfortably).
        const int woff = (y + RAD) * WP + (x + RAD);

        float acc = 0.0f;
        #pragma unroll
        for (int t = 0; t < 25; ++t) {
            // Uniform tap base (t is an unroll constant) -> SGPR/saddr;
            // per-thread woff -> VGPR offset. Weight stream is read-once:
            // non-temporal keeps L2 free for the reused hn working set.
            const float* __restrict__ bt = gwb + (size_t)t * (size_t)GW_HW;
            const float wv = __builtin_nontemporal_load(bt + woff);
            float sv;
            if (t == 12) {
                sv = h0v;                               // center tap reads h0
            } else {
                const int dy = t / 5;
                const int dx = t - dy * 5;
                // stack[t](y,x) = hn[y + 2 - dy][x + 2 - dx]
                sv = tile[(ly + 4 - dy) * LDSW + (lx + 4 - dx)];
            }
            acc = fmaf(wv, sv, acc);
        }
        // Write-once output stream -> non-temporal store.
        __builtin_nontemporal_store(acc, outb + (size_t)y * W + x);
    }
}

extern "C" void kernel_launch(void* const* d_in, const int* in_sizes, int n_in,
                              void* d_out, int out_size, void* d_ws, size_t ws_size,
                              hipStream_t stream)
{
    const float* gw = (const float*)d_in[0];   // (4, 25, 356, 1220) f32
    const float* hn = (const float*)d_in[1];   // (4, 1, 352, 1216) f32
    const float* h0 = (const float*)d_in[2];   // (4, 1, 352, 1216) f32
    float* out = (float*)d_out;                // (4, 1, 352, 1216) f32

    dim3 grid(W / TW, H / TH, BATCH);          // 19 x 22 x 4 = 1672 blocks
    dim3 block(NTHREADS);
    hipLaunchKernelGGL(cspn_25tap_kernel, grid, block, 0, stream, gw, hn, h0, out);
}